// kmeans_cluster_30047591202834
// MI455X (gfx1250) — compile-verified
//
#include <hip/hip_runtime.h>
#include <hip/hip_bf16.h>
#include <math.h>

// Problem constants (fixed by setup_inputs)
#define BN   8192      // batch
#define DD   768       // feature dim
#define KK   256       // clusters
#define EPS  1e-8f
#define ALPHA 0.2f
#define BETA  0.3f
#define NEG_INF_V (-10000.0f)
#define INV_WTEMP 20.0f   // 1/0.05

typedef float v2f __attribute__((ext_vector_type(2)));
typedef float v8f __attribute__((ext_vector_type(8)));

// ---------------------------------------------------------------------------
// Kernel 1: normalized centroids  cn[k][d] = c[k][d] / max(||c[k]||, eps)
// one block per centroid row
// ---------------------------------------------------------------------------
__global__ __launch_bounds__(256) void k_norm_centroid(const float* __restrict__ c,
                                                       float* __restrict__ cn) {
    const int row = blockIdx.x;
    const int t = threadIdx.x;
    __shared__ float sred[256];
    const float* src = c + (size_t)row * DD;
    float s = 0.f;
    for (int d = t; d < DD; d += 256) { float x = src[d]; s += x * x; }
    sred[t] = s; __syncthreads();
    for (int k = 128; k > 0; k >>= 1) { if (t < k) sred[t] += sred[t + k]; __syncthreads(); }
    const float inv = 1.0f / fmaxf(sqrtf(sred[0]), EPS);
    float* dst = cn + (size_t)row * DD;
    for (int d = t; d < DD; d += 256) dst[d] = src[d] * inv;
}

// ---------------------------------------------------------------------------
// Kernel 2: per-datapoint inverse norms  rinv[i] = 1 / max(||dp[i]||, eps)
// one block per datapoint row
// ---------------------------------------------------------------------------
__global__ __launch_bounds__(256) void k_row_invnorm(const float* __restrict__ dp,
                                                     float* __restrict__ rinv) {
    const int row = blockIdx.x;
    const int t = threadIdx.x;
    __shared__ float sred[256];
    const float* src = dp + (size_t)row * DD;
    float s = 0.f;
    for (int d = t; d < DD; d += 256) { float x = src[d]; s += x * x; }
    sred[t] = s; __syncthreads();
    for (int k = 128; k > 0; k >>= 1) { if (t < k) sred[t] += sred[t + k]; __syncthreads(); }
    if (t == 0) rinv[row] = 1.0f / fmaxf(sqrtf(sred[0]), EPS);
}

// ---------------------------------------------------------------------------
// Kernel 3: intra_cos_sim = (dp @ cn^T) * rinv[row]  via V_WMMA_F32_16X16X4_F32
// One wave32 computes a 16x16 tile of C. 8 waves/block share an M-strip.
// A layout (16x4 f32): lane<16 -> M=lane, K=k+{0,1}; lane>=16 -> K=k+{2,3}
// B layout (4x16 f32): lane<16 -> N=lane, K=k+{0,1}; lane>=16 -> K=k+{2,3}
// cn is [N][K] row-major, which is exactly the KxN operand order per lane.
// ---------------------------------------------------------------------------
__global__ __launch_bounds__(256) void k_gemm_wmma(const float* __restrict__ dp,
                                                   const float* __restrict__ cn,
                                                   const float* __restrict__ rinv,
                                                   float* __restrict__ out) {
    const int lane = threadIdx.x & 31;
    const int wave = threadIdx.x >> 5;
    const int m0 = blockIdx.x * 16;                 // 512 M-tiles
    const int n0 = (blockIdx.y * 8 + wave) * 16;    // 16 N-tiles total
    const int half = lane >> 4;
    const int idx  = lane & 15;

    const float* arow = dp + (size_t)(m0 + idx) * DD + 2 * half;
    const float* brow = cn + (size_t)(n0 + idx) * DD + 2 * half;

    v8f acc = {0.f, 0.f, 0.f, 0.f, 0.f, 0.f, 0.f, 0.f};
#pragma unroll 4
    for (int k = 0; k < DD; k += 4) {
        v2f a = *(const v2f*)(arow + k);
        v2f b = *(const v2f*)(brow + k);
        acc = __builtin_amdgcn_wmma_f32_16x16x4_f32(
            /*neg_a=*/false, a, /*neg_b=*/false, b,
            /*c_mod=*/(short)0, acc, /*reuse_a=*/false, /*reuse_b=*/false);
    }
    // C layout: acc[r] -> row m0 + half*8 + r, col n0 + idx
#pragma unroll
    for (int r = 0; r < 8; ++r) {
        const int row = m0 + half * 8 + r;
        out[(size_t)row * KK + n0 + idx] = acc[r] * rinv[row];
    }
}

// ---------------------------------------------------------------------------
// Kernel 4: per-row top-2 over intra_cos_sim (K=256), then gather centroid
// rows for dp_centroid (top-1) and hard_negative (top-2).
// Tie-break: lower index wins (matches argmax / lax.top_k).
// ---------------------------------------------------------------------------
__device__ __forceinline__ bool better(float v, int i, float w, int j) {
    return (v > w) || (v == w && i < j);
}

__global__ __launch_bounds__(256) void k_top2(const float* __restrict__ intra,
                                              const float* __restrict__ centroid,
                                              int* __restrict__ dpi,
                                              int* __restrict__ dpi2,
                                              float* __restrict__ o_cent,
                                              float* __restrict__ o_hneg) {
    const int i = blockIdx.x;
    const int t = threadIdx.x;
    __shared__ float s_v1[256], s_v2[256];
    __shared__ int   s_i1[256], s_i2[256];
    __shared__ int   s_best[2];

    s_v1[t] = intra[(size_t)i * KK + t];
    s_i1[t] = t;
    s_v2[t] = -INFINITY;
    s_i2[t] = 0x7fffffff;
    __syncthreads();

    for (int s = 128; s > 0; s >>= 1) {
        if (t < s) {
            float a1 = s_v1[t],   a2 = s_v2[t];
            int  ai1 = s_i1[t],  ai2 = s_i2[t];
            float b1 = s_v1[t+s], b2 = s_v2[t+s];
            int  bi1 = s_i1[t+s], bi2 = s_i2[t+s];
            float r1, r2; int ri1, ri2;
            if (better(a1, ai1, b1, bi1)) {
                r1 = a1; ri1 = ai1;
                if (better(a2, ai2, b1, bi1)) { r2 = a2; ri2 = ai2; }
                else                          { r2 = b1; ri2 = bi1; }
            } else {
                r1 = b1; ri1 = bi1;
                if (better(b2, bi2, a1, ai1)) { r2 = b2; ri2 = bi2; }
                else                          { r2 = a1; ri2 = ai1; }
            }
            s_v1[t] = r1; s_i1[t] = ri1; s_v2[t] = r2; s_i2[t] = ri2;
        }
        __syncthreads();
    }
    if (t == 0) {
        dpi[i]  = s_i1[0];
        dpi2[i] = s_i2[0];
        s_best[0] = s_i1[0];
        s_best[1] = s_i2[0];
    }
    __syncthreads();
    const int c1 = s_best[0], c2 = s_best[1];
    const float* src1 = centroid + (size_t)c1 * DD;
    const float* src2 = centroid + (size_t)c2 * DD;
    float* d1 = o_cent + (size_t)i * DD;
    float* d2 = o_hneg + (size_t)i * DD;
    for (int d = t; d < DD; d += 256) { d1[d] = src1[d]; d2[d] = src2[d]; }
}

// ---------------------------------------------------------------------------
// Kernel 5: fused per-row pass over batch_cos_sim (row resident in LDS):
//   dp_cluster[i][j], fn_loss partial (fixed-order reduce), masked softmax
//   weights[i][j] = exp(x - max) / sum / 0.05
// one block per row, row (32KB) staged in LDS so bcs is read from HBM once.
// ---------------------------------------------------------------------------
__global__ __launch_bounds__(256) void k_rowpass(const float* __restrict__ bcs,
                                                 const int* __restrict__ dpi,
                                                 float* __restrict__ o_clus,
                                                 float* __restrict__ o_wt,
                                                 float* __restrict__ lpart) {
    const int i = blockIdx.x;
    const int t = threadIdx.x;
    __shared__ float srow[BN];     // 32 KB of the 320 KB/WGP LDS
    __shared__ float sred[256];

    const int   myidx = dpi[i];
    const float pos   = bcs[(size_t)i * BN + i];
    const float* rsrc = bcs + (size_t)i * BN;
    float* cdst = o_clus + (size_t)i * BN;

    float lmax = -INFINITY;
    float lloss = 0.f;
    for (int j = t; j < BN; j += 256) {
        const float x = rsrc[j];
        srow[j] = x;
        const bool cl = (dpi[j] == myidx) && (j != i);
        cdst[j] = cl ? 1.0f : 0.0f;
        const float delta = cl ? (x - pos) : 0.0f;
        lloss += fmaxf(delta + ALPHA, 0.0f) + fmaxf(-delta - BETA, 0.0f);
        lmax = fmaxf(lmax, (j == i) ? NEG_INF_V : x);
    }
    // fixed-order loss reduction
    sred[t] = lloss; __syncthreads();
    for (int k = 128; k > 0; k >>= 1) { if (t < k) sred[t] += sred[t + k]; __syncthreads(); }
    if (t == 0) lpart[i] = sred[0];
    __syncthreads();
    // row max
    sred[t] = lmax; __syncthreads();
    for (int k = 128; k > 0; k >>= 1) { if (t < k) sred[t] = fmaxf(sred[t], sred[t + k]); __syncthreads(); }
    const float rmax = sred[0];
    __syncthreads();
    // exp sum from LDS
    float lsum = 0.f;
    for (int j = t; j < BN; j += 256) {
        const float v = (j == i) ? NEG_INF_V : srow[j];
        lsum += __expf(v - rmax);
    }
    sred[t] = lsum; __syncthreads();
    for (int k = 128; k > 0; k >>= 1) { if (t < k) sred[t] += sred[t + k]; __syncthreads(); }
    const float scale = INV_WTEMP / sred[0];
    __syncthreads();
    // weights (scalar stores: weights base is only 4B-aligned in d_out)
    float* wdst = o_wt + (size_t)i * BN;
    for (int j = t; j < BN; j += 256) {
        const float v = (j == i) ? NEG_INF_V : srow[j];
        wdst[j] = __expf(v - rmax) * scale;
    }
}

// ---------------------------------------------------------------------------
// Kernel 6: deterministic final loss reduction: mean over B*B
// ---------------------------------------------------------------------------
__global__ __launch_bounds__(256) void k_loss_final(const float* __restrict__ lpart,
                                                    float* __restrict__ out) {
    __shared__ float sred[256];
    const int t = threadIdx.x;
    float s = 0.f;
    for (int i = t; i < BN; i += 256) s += lpart[i];
    sred[t] = s; __syncthreads();
    for (int k = 128; k > 0; k >>= 1) { if (t < k) sred[t] += sred[t + k]; __syncthreads(); }
    if (t == 0) out[0] = sred[0] / ((float)BN * (float)BN);
}

// ---------------------------------------------------------------------------
extern "C" void kernel_launch(void* const* d_in, const int* in_sizes, int n_in,
                              void* d_out, int out_size, void* d_ws, size_t ws_size,
                              hipStream_t stream) {
    const float* dp  = (const float*)d_in[0];   // [B, D]
    const float* cen = (const float*)d_in[1];   // [K, D]
    const float* bcs = (const float*)d_in[2];   // [B, B]

    // output layout (flat f32, return order)
    float* o       = (float*)d_out;
    float* o_intra = o;                                   // B*K
    float* o_clus  = o + (size_t)BN * KK;                 // B*B
    float* o_cent  = o_clus + (size_t)BN * BN;            // B*D
    float* o_hneg  = o_cent + (size_t)BN * DD;            // B*D
    float* o_loss  = o_hneg + (size_t)BN * DD;            // 1
    float* o_wt    = o_loss + 1;                          // B*B

    // workspace layout (~0.9 MB)
    float* cn    = (float*)d_ws;                          // K*D
    float* rinv  = cn + (size_t)KK * DD;                  // B
    int*   dpi   = (int*)(rinv + BN);                     // B
    int*   dpi2  = dpi + BN;                              // B
    float* lpart = (float*)(dpi2 + BN);                   // B

    k_norm_centroid<<<KK, 256, 0, stream>>>(cen, cn);
    k_row_invnorm<<<BN, 256, 0, stream>>>(dp, rinv);
    k_gemm_wmma<<<dim3(BN / 16, 2), 256, 0, stream>>>(dp, cn, rinv, o_intra);
    k_top2<<<BN, 256, 0, stream>>>(o_intra, cen, dpi, dpi2, o_cent, o_hneg);
    k_rowpass<<<BN, 256, 0, stream>>>(bcs, dpi, o_clus, o_wt, lpart);
    k_loss_final<<<1, 256, 0, stream>>>(lpart, o_loss);
}